// _NonLocalLayer_51762945851646
// MI455X (gfx1250) — compile-verified
//
#include <hip/hip_runtime.h>

typedef float v2f __attribute__((ext_vector_type(2)));
typedef float v4f __attribute__((ext_vector_type(4)));
typedef float v8f __attribute__((ext_vector_type(8)));

namespace {
constexpr int kInDim  = 256;
constexpr int kFF     = 512;
constexpr int kOutDim = 256;
constexpr int kB      = 4;
constexpr int kN      = 4096;               // 16*16*16 spatial positions per batch
constexpr float kInvTot = 1.0f / 16384.0f;  // 1 / (B * N)
constexpr float kEps  = 1e-5f;

constexpr int TPB        = 256;  // 8 waves (wave32)
constexpr int TILE_N     = 64;   // columns per workgroup
constexpr int TILE_M     = 128;  // rows per workgroup = 8 waves * 16
constexpr int KCHUNK     = 128;  // K rows staged in LDS per chunk
constexpr int KPAIRS     = KCHUNK / 2;
// Interleaved layout: Xs[kpair*STRIDE + 2*col + parity] = X[2*kpair + parity][col].
// STRIDE = 2*TILE_N + 32 so (kpair+1) row is 32 banks away -> the two half-waves
// (which read kpair and kpair+1) hit disjoint bank halves on ds_load_b64.
constexpr int XS_STRIDE2 = 2 * TILE_N + 32;  // 160 floats
}

// Y = relu(W @ X + bias)                        (MODE 0)
// Y = resid + relu(W @ X + bias), + BN partials (MODE 1)
// W: (odim x CDIM) row-major.  X,Y,resid: (B, dim, kN), row stride kN.
template <int CDIM, int MODE>
__launch_bounds__(TPB)
__global__ void gemm_wmma(const float* __restrict__ W,
                          const float* __restrict__ X,
                          const float* __restrict__ bias,
                          const float* __restrict__ resid,
                          float* __restrict__ Y,
                          float* __restrict__ gsum,
                          float* __restrict__ gsumsq,
                          int odim) {
  __shared__ float Xs[KPAIRS * XS_STRIDE2];   // 40 KB

  const int tid  = threadIdx.x;
  const int wave = tid >> 5;
  const int lane = tid & 31;
  const int l16  = lane & 15;
  const int kh   = lane >> 4;                 // 0: K={0,1}, 1: K={2,3} half of WMMA K
  const int b    = blockIdx.z;
  const int n0   = blockIdx.x * TILE_N;
  const int m0   = blockIdx.y * TILE_M + wave * 16;

  const float* __restrict__ Xb = X + (size_t)b * CDIM * kN;
  // A-operand row for this lane: W[m0 + l16][.. + 2*kh]
  const float* __restrict__ Wrow = W + (size_t)(m0 + l16) * CDIM + 2 * kh;

  v8f acc0 = {}, acc1 = {}, acc2 = {}, acc3 = {};

  for (int kb = 0; kb < CDIM; kb += KCHUNK) {
    __syncthreads();
    // Stage X[kb:kb+128, n0:n0+64] -> LDS, K-pair interleaved.
    // Each iteration: two coalesced float4 global loads (rows k, k+1), two 16B LDS stores.
#pragma unroll
    for (int it = 0; it < (KPAIRS * (TILE_N / 4)) / TPB; ++it) {
      const int idx = tid + it * TPB;
      const int kp  = idx >> 4;            // 0..63
      const int c   = (idx & 15) << 2;     // 0,4,..,60
      const float* g = Xb + (size_t)(kb + 2 * kp) * kN + n0 + c;
      v4f a = *(const v4f*)g;              // X[2kp][c..c+3]
      v4f d = *(const v4f*)(g + kN);       // X[2kp+1][c..c+3]
      v4f s0, s1;
      s0.x = a.x; s0.y = d.x; s0.z = a.y; s0.w = d.y;
      s1.x = a.z; s1.y = d.z; s1.z = a.w; s1.w = d.w;
      float* p = &Xs[kp * XS_STRIDE2 + 2 * c];
      *(v4f*)(p + 0) = s0;
      *(v4f*)(p + 4) = s1;
    }
    __syncthreads();

#pragma unroll 8
    for (int k = 0; k < KCHUNK; k += 4) {
      // A: 16x4 f32 tile of W (lane<16 -> K0/K1, lane>=16 -> K2/K3)
      v2f a = *(const v2f*)(Wrow + kb + k);
      // B operands: one aligned ds_load_b64 each, no repacking.
      const float* bp = &Xs[((k >> 1) + kh) * XS_STRIDE2 + 2 * l16];
      v2f b0 = *(const v2f*)(bp + 0);
      v2f b1 = *(const v2f*)(bp + 32);
      v2f b2 = *(const v2f*)(bp + 64);
      v2f b3 = *(const v2f*)(bp + 96);
      acc0 = __builtin_amdgcn_wmma_f32_16x16x4_f32(false, a, false, b0, (short)0, acc0, false, false);
      acc1 = __builtin_amdgcn_wmma_f32_16x16x4_f32(false, a, false, b1, (short)0, acc1, false, false);
      acc2 = __builtin_amdgcn_wmma_f32_16x16x4_f32(false, a, false, b2, (short)0, acc2, false, false);
      acc3 = __builtin_amdgcn_wmma_f32_16x16x4_f32(false, a, false, b3, (short)0, acc3, false, false);
    }
  }

  // Epilogue. D layout: VGPR r -> row (m0 + r + 8*kh), lanes l16 -> column n within tile.
  float bias_r[8];
#pragma unroll
  for (int r = 0; r < 8; ++r) bias_r[r] = bias[m0 + r + 8 * kh];

  float* __restrict__ Yb = Y + (size_t)b * odim * kN;

  if (MODE == 0) {
#pragma unroll
    for (int r = 0; r < 8; ++r) {
      const int row = m0 + r + 8 * kh;
      float* yr = Yb + (size_t)row * kN + n0 + l16;
      yr[0]  = fmaxf(acc0[r] + bias_r[r], 0.0f);
      yr[16] = fmaxf(acc1[r] + bias_r[r], 0.0f);
      yr[32] = fmaxf(acc2[r] + bias_r[r], 0.0f);
      yr[48] = fmaxf(acc3[r] + bias_r[r], 0.0f);
    }
  } else {
    const float* __restrict__ Rb = resid + (size_t)b * odim * kN;
#pragma unroll
    for (int r = 0; r < 8; ++r) {
      const int row = m0 + r + 8 * kh;
      const float* rr = Rb + (size_t)row * kN + n0 + l16;
      float*       yr = Yb + (size_t)row * kN + n0 + l16;
      const float v0 = rr[0]  + fmaxf(acc0[r] + bias_r[r], 0.0f);
      const float v1 = rr[16] + fmaxf(acc1[r] + bias_r[r], 0.0f);
      const float v2 = rr[32] + fmaxf(acc2[r] + bias_r[r], 0.0f);
      const float v3 = rr[48] + fmaxf(acc3[r] + bias_r[r], 0.0f);
      yr[0] = v0; yr[16] = v1; yr[32] = v2; yr[48] = v3;
      float s = v0 + v1 + v2 + v3;
      float q = v0 * v0 + v1 * v1 + v2 * v2 + v3 * v3;
      // reduce across the 16 lanes that share this channel row
#pragma unroll
      for (int off = 8; off >= 1; off >>= 1) {
        s += __shfl_xor(s, off, 16);
        q += __shfl_xor(q, off, 16);
      }
      if (l16 == 0) {
        atomicAdd(&gsum[row], s);
        atomicAdd(&gsumsq[row], q);
      }
    }
  }
}

__global__ void zero_kernel(float* __restrict__ p, int n) {
  const int i = blockIdx.x * blockDim.x + threadIdx.x;
  if (i < n) p[i] = 0.0f;
}

__global__ void bn_stats_kernel(const float* __restrict__ gsum,
                                const float* __restrict__ gsumsq,
                                float* __restrict__ mean,
                                float* __restrict__ rstd) {
  const int o = threadIdx.x;
  const float m = gsum[o] * kInvTot;
  const float v = gsumsq[o] * kInvTot - m * m;   // biased var, as jnp.var
  mean[o] = m;
  rstd[o] = rsqrtf(v + kEps);
}

__global__ void bn_apply_kernel(const float* __restrict__ res,
                                const float* __restrict__ mean,
                                const float* __restrict__ rstd,
                                const float* __restrict__ gamma,
                                const float* __restrict__ beta,
                                float* __restrict__ out) {
  const size_t i4 = (size_t)blockIdx.x * blockDim.x + threadIdx.x;
  const size_t e  = i4 * 4;                    // 4 floats stay inside one channel row (kN % 4 == 0)
  const int o = (int)((e >> 12) & 255);        // (e / 4096) % 256
  const float g  = gamma[o] * rstd[o];
  const float bb = beta[o] - mean[o] * g;
  v4f r = *(const v4f*)(res + e);
  v4f y;
  y.x = r.x * g + bb; y.y = r.y * g + bb; y.z = r.z * g + bb; y.w = r.w * g + bb;
  *(v4f*)(out + e) = y;
}

extern "C" void kernel_launch(void* const* d_in, const int* in_sizes, int n_in,
                              void* d_out, int out_size, void* d_ws, size_t ws_size,
                              hipStream_t stream) {
  (void)in_sizes; (void)n_in; (void)out_size; (void)ws_size;
  // setup_inputs order: inputs, Wk, bk, Wq, bq, Wv, bv, Wo, bo, gamma, beta
  // K/Q/softmax are dead: rowsum(softmax) == 1, so Vagg == V.
  const float* x     = (const float*)d_in[0];
  const float* Wv    = (const float*)d_in[5];
  const float* bv    = (const float*)d_in[6];
  const float* Wo    = (const float*)d_in[7];
  const float* bo    = (const float*)d_in[8];
  const float* gamma = (const float*)d_in[9];
  const float* beta  = (const float*)d_in[10];

  float* ws   = (float*)d_ws;
  float* V1   = ws;                                   // B*FF*N      = 8388608 f32
  float* res  = V1 + (size_t)kB * kFF * kN;           // B*OUT*N     = 4194304 f32
  float* gsum = res + (size_t)kB * kOutDim * kN;      // 256
  float* gsq  = gsum + 256;                           // 256
  float* mean = gsq + 256;                            // 256
  float* rstd = mean + 256;                           // 256
  float* outp = (float*)d_out;

  zero_kernel<<<2, 256, 0, stream>>>(gsum, 512);      // sum + sumsq

  dim3 blk(TPB);
  dim3 g1(kN / TILE_N, kFF / TILE_M, kB);             // (64, 4, 4)
  gemm_wmma<kInDim, 0><<<g1, blk, 0, stream>>>(Wv, x, bv, nullptr, V1, nullptr, nullptr, kFF);

  dim3 g2(kN / TILE_N, kOutDim / TILE_M, kB);         // (64, 2, 4)
  gemm_wmma<kFF, 1><<<g2, blk, 0, stream>>>(Wo, V1, bo, x, res, gsum, gsq, kOutDim);

  bn_stats_kernel<<<1, 256, 0, stream>>>(gsum, gsq, mean, rstd);

  const int nvec4 = (kB * kOutDim * kN) / 4;          // 1048576
  bn_apply_kernel<<<nvec4 / TPB, TPB, 0, stream>>>(res, mean, rstd, gamma, beta, outp);
}